// NLinear_9637906612431
// MI455X (gfx1250) — compile-verified
//
#include <hip/hip_runtime.h>

// NLinear on MI455X (gfx1250):
//   Stage 1: memory-bound depthwise contraction over w (1 GiB of x @ 23.3 TB/s
//            -> ~46us floor). Coalesced b128 loads, register accumulation,
//            LDS reduction. global_prefetch_b8 for streaming.
//   Stage 2: Y(2048x64) @ W2(64x1) via V_WMMA_F32_16X16X4_F32 (16 chained
//            K=4 steps per 16-row tile). B operand built branchlessly
//            (mask multiply) so EXEC stays all-ones and no saveexec noise.

#define B_N   2048
#define WIN_N 2048
#define E_N   64
#define BPB   4      // batch rows per block in stage 1

typedef __attribute__((ext_vector_type(2))) float v2f;
typedef __attribute__((ext_vector_type(4))) float v4f;
typedef __attribute__((ext_vector_type(8))) float v8f;

__global__ __launch_bounds__(256) void nlinear_stage1(
    const float* __restrict__ x,   // [B, WIN, E]
    const float* __restrict__ W1,  // [E, WIN]
    const float* __restrict__ b1,  // [E]
    float* __restrict__ Y)         // [B, E] (workspace)
{
    __shared__ float red[16 * E_N];

    const int t   = threadIdx.x;
    const int eq  = t & 15;   // channel quad: e = 4*eq + j
    const int wsl = t >> 4;   // w slot 0..15
    const int b0  = blockIdx.x * BPB;

    const v4f* __restrict__ x4 = (const v4f*)x;
    const v4f zero = {0.f, 0.f, 0.f, 0.f};

    v4f L[BPB], acc[BPB];
#pragma unroll
    for (int i = 0; i < BPB; ++i) {
        // seq_last = x[b, WIN-1, e..e+3]
        L[i] = x4[((size_t)(b0 + i) * WIN_N + (WIN_N - 1)) * (E_N / 4) + eq];
        acc[i] = zero;
    }

    const float* __restrict__ w1r = W1 + (size_t)(4 * eq) * WIN_N;

    for (int w = wsl; w < WIN_N; w += 16) {
        const float wa = w1r[w];
        const float wb = w1r[WIN_N + w];
        const float wc = w1r[2 * WIN_N + w];
        const float wd = w1r[3 * WIN_N + w];

        // stream-ahead prefetch of x (gfx1250 global_prefetch_b8)
        if (w + 128 < WIN_N)
            __builtin_prefetch(
                (const void*)&x4[((size_t)b0 * WIN_N + (w + 128)) * (E_N / 4) + eq], 0, 1);

#pragma unroll
        for (int i = 0; i < BPB; ++i) {
            v4f xv = x4[((size_t)(b0 + i) * WIN_N + w) * (E_N / 4) + eq];
            acc[i].x += (xv.x - L[i].x) * wa;
            acc[i].y += (xv.y - L[i].y) * wb;
            acc[i].z += (xv.z - L[i].z) * wc;
            acc[i].w += (xv.w - L[i].w) * wd;
        }
    }

    // Reduce the 16 w-slots per channel, one batch row at a time.
    for (int i = 0; i < BPB; ++i) {
        __syncthreads();
        *(v4f*)&red[wsl * E_N + 4 * eq] = acc[i];
        __syncthreads();
        if (t < E_N) {
            float s = 0.f;
#pragma unroll
            for (int k = 0; k < 16; ++k) s += red[k * E_N + t];
            const float Le = x[((size_t)(b0 + i) * WIN_N + (WIN_N - 1)) * E_N + t];
            Y[(b0 + i) * E_N + t] = s + b1[t] + Le;
        }
    }
}

__global__ __launch_bounds__(32) void nlinear_stage2(
    const float* __restrict__ Y,   // [B, E]
    const float* __restrict__ W2,  // [E]
    const float* __restrict__ b2,  // [1]
    float* __restrict__ out)       // [B]
{
    const int lane = threadIdx.x;      // 0..31, all active (EXEC all-ones for WMMA)
    const int half = lane >> 4;        // 0: K={0,1}, 1: K={2,3} (ISA A/B 16x4 layout)
    const int m    = lane & 15;        // row (A) / column (B)
    const int b0   = blockIdx.x * 16;

    // Branchless column mask: only column n==0 of B carries W2.
    const float msk = (m == 0) ? 1.f : 0.f;

    const v2f* __restrict__ Yv  = (const v2f*)Y;
    const v2f* __restrict__ W2v = (const v2f*)W2;

    v8f c = {0.f, 0.f, 0.f, 0.f, 0.f, 0.f, 0.f, 0.f};

#pragma unroll
    for (int j = 0; j < 16; ++j) {     // K = 64 in 16 steps of 4
        // K-pair for this lane half: k0 = 4j + 2*half (even -> one b64 load each)
        v2f a  = Yv[((b0 + m) * E_N + 4 * j) / 2 + half];
        v2f wv = W2v[2 * j + half];
        v2f bm;
        bm.x = wv.x * msk;
        bm.y = wv.y * msk;
        c = __builtin_amdgcn_wmma_f32_16x16x4_f32(
                /*neg_a=*/false, a, /*neg_b=*/false, bm,
                /*c_mod=*/(short)0, c, /*reuse_a=*/false, /*reuse_b=*/false);
    }

    const float bias = b2[0];
    // D column 0: rows 0..7 in lane 0 (VGPRs 0..7), rows 8..15 in lane 16.
    if (m == 0) {
        const int rb = half * 8;
#pragma unroll
        for (int r = 0; r < 8; ++r)
            out[b0 + rb + r] = c[r] + bias;
    }
}

extern "C" void kernel_launch(void* const* d_in, const int* in_sizes, int n_in,
                              void* d_out, int out_size, void* d_ws, size_t ws_size,
                              hipStream_t stream) {
    const float* x  = (const float*)d_in[0];
    const float* W1 = (const float*)d_in[1];
    const float* b1 = (const float*)d_in[2];
    const float* W2 = (const float*)d_in[3];
    const float* b2 = (const float*)d_in[4];
    float* out = (float*)d_out;
    float* Y   = (float*)d_ws;   // B_N * E_N floats = 512 KB scratch

    nlinear_stage1<<<B_N / BPB, 256, 0, stream>>>(x, W1, b1, Y);
    nlinear_stage2<<<B_N / 16, 32, 0, stream>>>(Y, W2, b2, out);
    (void)in_sizes; (void)n_in; (void)out_size; (void)ws_size;
}